// GNNDecoder_80032420594393
// MI455X (gfx1250) — compile-verified
//
#include <hip/hip_runtime.h>

#define N_V 500000
#define N_E 16000000

typedef __attribute__((ext_vector_type(2))) float v2f;
typedef __attribute__((ext_vector_type(8))) float v8f;

// ---------------------------------------------------------------------------
// d_ws layout (floats):
//   [0, N_V*3)          : segment-sum accumulator h
//   [N_V*3, N_V*3+18)   : fused M = We @ Wh   (6x3 row-major)
//   [N_V*3+18, +21)     : fused c = be @ Wh   (3)
// ---------------------------------------------------------------------------

__global__ __launch_bounds__(256) void gnn_zero_acc(float* __restrict__ h) {
    int i = blockIdx.x * blockDim.x + threadIdx.x;
    int stride = gridDim.x * blockDim.x;
    float4* h4 = (float4*)h;
    const int n4 = (N_V * 3) / 4;            // 375000 exactly
    for (int j = i; j < n4; j += stride) h4[j] = make_float4(0.f, 0.f, 0.f, 0.f);
}

__global__ void gnn_prep(const float* __restrict__ We, const float* __restrict__ be,
                         const float* __restrict__ Wh, float* __restrict__ Mc) {
    if (threadIdx.x == 0 && blockIdx.x == 0) {
        // M[k][j] = We[k][0]*Wh[0][j] + We[k][1]*Wh[1][j]   (We: 6x2, Wh: 2x3)
        for (int k = 0; k < 6; ++k)
            for (int j = 0; j < 3; ++j)
                Mc[k * 3 + j] = fmaf(We[k * 2 + 0], Wh[0 * 3 + j],
                                     We[k * 2 + 1] * Wh[1 * 3 + j]);
        for (int j = 0; j < 3; ++j)
            Mc[18 + j] = fmaf(be[0], Wh[0 * 3 + j], be[1] * Wh[1 * 3 + j]);
    }
}

__device__ __forceinline__ void scatter_add(float* p, float v) {
    // Non-returning relaxed agent-scope fadd -> GLOBAL_ATOMIC_ADD_F32 (no RET),
    // executed at L2 where the 6MB accumulator lives.
    (void)__hip_atomic_fetch_add(p, v, __ATOMIC_RELAXED, __HIP_MEMORY_SCOPE_AGENT);
}

// One pass over the 192MB edge/prob stream; gathers + atomics stay in L2
// (vertex table and accumulator are 6MB each, global L2 is 192MB).
// Two edges per iteration: edge stream read as b128, prob stream as b64.
__global__ __launch_bounds__(256) void gnn_edges(
        const float* __restrict__ vtx, const float* __restrict__ prob,
        const int*   __restrict__ edges, const float* __restrict__ Mc,
        const float* __restrict__ bh, float* __restrict__ hacc) {
    float M[18];
#pragma unroll
    for (int t = 0; t < 18; ++t) M[t] = Mc[t];
    const float c0 = Mc[18], c1 = Mc[19], c2 = Mc[20];
    const float b0 = bh[0], b1 = bh[1], b2 = bh[2];

    const int4*   ep = (const int4*)edges;    // 2 edges per element
    const float2* pp = (const float2*)prob;   // 2 probs per element
    int tid = blockIdx.x * blockDim.x + threadIdx.x;
    int stride = gridDim.x * blockDim.x;
    const int nPairs = N_E / 2;               // 8M, exact

    for (int i = tid; i < nPairs; i += stride) {
        int4   e2 = ep[i];                    // global_load_b128
        float2 p2 = pp[i];                    // global_load_b64

#pragma unroll
        for (int k = 0; k < 2; ++k) {
            int src = (k == 0) ? e2.x : e2.z;
            int tgt = (k == 0) ? e2.y : e2.w;
            float p = (k == 0) ? p2.x : p2.y;
            int s3 = src * 3, t3 = tgt * 3;
            float x0 = vtx[s3 + 0], x1 = vtx[s3 + 1], x2 = vtx[s3 + 2];
            float x3 = vtx[t3 + 0], x4 = vtx[t3 + 1], x5 = vtx[t3 + 2];

            float m0 = fmaf(x0, M[0], fmaf(x1, M[3], fmaf(x2, M[6],
                       fmaf(x3, M[9], fmaf(x4, M[12], fmaf(x5, M[15], c0))))));
            float m1 = fmaf(x0, M[1], fmaf(x1, M[4], fmaf(x2, M[7],
                       fmaf(x3, M[10], fmaf(x4, M[13], fmaf(x5, M[16], c1))))));
            float m2 = fmaf(x0, M[2], fmaf(x1, M[5], fmaf(x2, M[8],
                       fmaf(x3, M[11], fmaf(x4, M[14], fmaf(x5, M[17], c2))))));
            m0 = fmaf(p, m0, b0); m1 = fmaf(p, m1, b1); m2 = fmaf(p, m2, b2);

            scatter_add(&hacc[t3 + 0], m0);
            scatter_add(&hacc[t3 + 1], m1);
            scatter_add(&hacc[t3 + 2], m2);
        }
    }
}

// Epilogue: out = vertices + h @ Wv + bv  via V_WMMA_F32_16X16X4_F32.
// One wave handles 16 rows. K padded 3->4, N padded 3->16 with zeros.
__global__ __launch_bounds__(256) void gnn_vertices(
        const float* __restrict__ vtx, const float* __restrict__ hacc,
        const float* __restrict__ Wv, const float* __restrict__ bv,
        float* __restrict__ out) {
    const int NT = N_V / 16;                           // 31250 tiles, exact
    int wave = (blockIdx.x * blockDim.x + threadIdx.x) >> 5;   // wave-uniform
    if (wave >= NT) return;                            // uniform branch: EXEC all-1s below

    int lane = threadIdx.x & 31;
    int half = lane >> 4;        // 0: K=0,1  |  1: K=2,3 (A); K rows 0,1 / 2,3 (B)
    int m    = lane & 15;        // A: row-in-tile; B/D: column
    int r0   = wave * 16;

    // A (16x4): lanes 0-15 VGPR0/1 = K0/K1; lanes 16-31 VGPR0/1 = K2/K3(pad=0)
    v2f a;
    {
        int row = r0 + m;
        if (half == 0) { a.x = hacc[row * 3 + 0]; a.y = hacc[row * 3 + 1]; }
        else           { a.x = hacc[row * 3 + 2]; a.y = 0.f; }
    }
    // B (4x16): VGPR0/1 = rows K0/K1 (lanes 0-15), rows K2/K3 (lanes 16-31), col = m
    v2f b; b.x = 0.f; b.y = 0.f;
    if (m < 3) {
        if (half == 0) { b.x = Wv[0 * 3 + m]; b.y = Wv[1 * 3 + m]; }
        else           { b.x = Wv[2 * 3 + m]; /* K=3 row stays 0 */ }
    }

    v8f c = {};
    v8f d = __builtin_amdgcn_wmma_f32_16x16x4_f32(
        /*neg_a=*/false, a, /*neg_b=*/false, b,
        /*c_mod=*/(short)0, c, /*reuse_a=*/false, /*reuse_b=*/false);

    // D (16x16): VGPR v holds row (v + 8*half), col m
    if (m < 3) {
        float bvn = bv[m];
#pragma unroll
        for (int v = 0; v < 8; ++v) {
            int rr = r0 + half * 8 + v;
            out[rr * 3 + m] = vtx[rr * 3 + m] + d[v] + bvn;
        }
    }
}

extern "C" void kernel_launch(void* const* d_in, const int* in_sizes, int n_in,
                              void* d_out, int out_size, void* d_ws, size_t ws_size,
                              hipStream_t stream) {
    const float* vtx   = (const float*)d_in[0];
    const float* prob  = (const float*)d_in[1];
    const int*   edges = (const int*)  d_in[2];
    const float* We    = (const float*)d_in[3];
    const float* be    = (const float*)d_in[4];
    const float* Wh    = (const float*)d_in[5];
    const float* bh    = (const float*)d_in[6];
    const float* Wv    = (const float*)d_in[7];
    const float* bv    = (const float*)d_in[8];
    float* out  = (float*)d_out;
    float* hacc = (float*)d_ws;
    float* Mc   = hacc + (size_t)N_V * 3;

    gnn_zero_acc<<<1024, 256, 0, stream>>>(hacc);
    gnn_prep<<<1, 32, 0, stream>>>(We, be, Wh, Mc);
    gnn_edges<<<8192, 256, 0, stream>>>(vtx, prob, edges, Mc, bh, hacc);
    const int vblocks = (N_V / 16 + 7) / 8;            // 8 waves per 256-thread block
    gnn_vertices<<<vblocks, 256, 0, stream>>>(vtx, hacc, Wv, bv, out);
}